// MultiHeadRecurrentAttentionEUNN_59373627900141
// MI455X (gfx1250) — compile-verified
//
#include <hip/hip_runtime.h>
#include <math.h>

// ---------------------------------------------------------------------------
// MultiHeadRecurrentAttentionEUNN for MI455X (gfx1250, wave32, WMMA)
//
// Pipeline:
//   1) gemm_qkv   : q/k/v = x @ W + b   (f32 WMMA 16x16x4), head-major out
//   2) logits     : q·k^T / 8           (f32 WMMA 16x16x4) -> d_out attn region
//   3) softmax    : per-row softmax (wave32 shfl reduction), also writes attn^T
//   4) gemm_vw    : vW = v @ (Win_re, Win_im)  (f32 WMMA), interleaved re/im
//   5) scan       : 256-step EUNN recurrence, state in VGPRs; per-step vW row
//                   double-buffered in LDS via GLOBAL_LOAD_ASYNC_TO_LDS_B128
//   6) gemm_out   : out = real(hT) @ Wo + bo   (f32 WMMA)
// ---------------------------------------------------------------------------

typedef __attribute__((ext_vector_type(2))) float v2f;
typedef __attribute__((ext_vector_type(8))) float v8f;
typedef __attribute__((ext_vector_type(4))) int   v4i;

typedef __attribute__((address_space(1))) v4i g_v4i;  // global
typedef __attribute__((address_space(3))) v4i l_v4i;  // LDS

#define D_MODEL   512
#define NUM_HEADS 8
#define D_QKV     64
#define B_SZ      2
#define L_SEQ     256
#define ROWS      (B_SZ * L_SEQ)     /* 512 */
#define BH        (B_SZ * NUM_HEADS) /* 16 */
#define VW_ROW    (2 * D_QKV)        /* 128 floats = 512 B interleaved re/im */

// V_WMMA_F32_16X16X4_F32 : D(16x16 f32) = A(16x4 f32) * B(4x16 f32) + C
// Fragment layout (per CDNA5 ISA 7.12.2):
//   A: lane l -> row m = l&15 ; reg0 = A[m][kb], reg1 = A[m][kb+1], kb = (l>>4)*2
//   B: lane l -> col n = l&15 ; reg0 = B[kb][n], reg1 = B[kb+1][n]
//   C/D: reg r, lane l -> M = r + 8*(l>>4), N = l&15
__device__ __forceinline__ v8f wmma4(v2f a, v2f b, v8f c) {
  return __builtin_amdgcn_wmma_f32_16x16x4_f32(
      /*neg_a=*/false, a, /*neg_b=*/false, b,
      /*c_mod=*/(short)0, c, /*reuse_a=*/false, /*reuse_b=*/false);
}

// ---- CDNA5 async global->LDS copy (ASYNCcnt-tracked DMA) ------------------
// Builtin exists on this toolchain (confirmed by round-2 diagnostic): param 0
// is a global-AS pointer to int __vector(4); param 1 the LDS-AS counterpart.
__device__ __forceinline__ void async_load_lds_b128(const float* g, float* l) {
#if __has_builtin(__builtin_amdgcn_global_load_async_to_lds_b128)
  __builtin_amdgcn_global_load_async_to_lds_b128(
      (g_v4i*)g, (l_v4i*)l, /*imm offset=*/0, /*cpol=*/0);
#else
  unsigned ldsoff = (unsigned)(unsigned long long)l;  // low 32b = LDS byte offset
  unsigned long long ga = (unsigned long long)g;
  asm volatile("global_load_async_to_lds_b128 %0, %1, off"
               :: "v"(ldsoff), "v"(ga) : "memory");
#endif
}

__device__ __forceinline__ void wait_async0() {
#if __has_builtin(__builtin_amdgcn_s_wait_asynccnt)
  __builtin_amdgcn_s_wait_asynccnt(0);
#else
  asm volatile("s_wait_asynccnt 0x0" ::: "memory");
#endif
}

// ---------------------------------------------------------------------------
// 1) QKV projection: rows = B*L = 512, cols = 512, K = 512.
//    One 16x16 output tile per wave; 3 matrices * 32*32 tiles = 3072 waves.
//    Output head-major: OUT[((b*8+h)*256 + l)*64 + d]
// ---------------------------------------------------------------------------
__global__ void gemm_qkv_kernel(const float* __restrict__ xq,
                                const float* __restrict__ xk,
                                const float* __restrict__ xv,
                                const float* __restrict__ Wq,
                                const float* __restrict__ Wk,
                                const float* __restrict__ Wv,
                                const float* __restrict__ bq,
                                const float* __restrict__ bk,
                                const float* __restrict__ bv,
                                float* __restrict__ qh,
                                float* __restrict__ kh,
                                float* __restrict__ vh) {
  const int wave = (blockIdx.x * blockDim.x + threadIdx.x) >> 5;
  const int lane = threadIdx.x & 31;
  const int mat  = wave / (32 * 32);
  const int t2   = wave % (32 * 32);
  const int tr = t2 / 32, tc = t2 % 32;

  const float* X    = (mat == 0) ? xq : (mat == 1) ? xk : xv;
  const float* W    = (mat == 0) ? Wq : (mat == 1) ? Wk : Wv;
  const float* bias = (mat == 0) ? bq : (mat == 1) ? bk : bv;
  float*       OUT  = (mat == 0) ? qh : (mat == 1) ? kh : vh;

  const int m15 = lane & 15;
  const int kb  = (lane >> 4) << 1;
  const int rowBase = tr * 16, colBase = tc * 16;
  const float* Arow = X + (size_t)(rowBase + m15) * D_MODEL;

  v8f acc = {0.f, 0.f, 0.f, 0.f, 0.f, 0.f, 0.f, 0.f};
  for (int k0 = 0; k0 < D_MODEL; k0 += 4) {
    v2f a;  a.x  = Arow[k0 + kb];
            a.y  = Arow[k0 + kb + 1];
    v2f bm; bm.x = W[(size_t)(k0 + kb)     * D_MODEL + colBase + m15];
            bm.y = W[(size_t)(k0 + kb + 1) * D_MODEL + colBase + m15];
    acc = wmma4(a, bm, acc);
  }

  const int col = colBase + m15;
  const int h = col / D_QKV, d = col % D_QKV;
  const float bval = bias[col];
#pragma unroll
  for (int r = 0; r < 8; ++r) {
    const int m   = r + ((lane >> 4) << 3);
    const int row = rowBase + m;             // 0..511
    const int b   = row / L_SEQ, l = row % L_SEQ;
    OUT[((size_t)(b * NUM_HEADS + h) * L_SEQ + l) * D_QKV + d] = acc[r] + bval;
  }
}

// ---------------------------------------------------------------------------
// 2) logits[b,h,q,k] = (q_h[b,h,q,:] . k_h[b,h,k,:]) / sqrt(64)
//    16 bh * 16*16 tiles = 4096 waves. B-matrix is K^T, so the B fragment
//    loads contiguous pairs from k_h rows.
// ---------------------------------------------------------------------------
__global__ void logits_kernel(const float* __restrict__ qh,
                              const float* __restrict__ kh,
                              float* __restrict__ attn) {
  const int wave = (blockIdx.x * blockDim.x + threadIdx.x) >> 5;
  const int lane = threadIdx.x & 31;
  const int bh = wave / (16 * 16);
  const int t2 = wave % (16 * 16);
  const int tr = t2 / 16, tc = t2 % 16;

  const float* Q = qh + (size_t)bh * L_SEQ * D_QKV;
  const float* K = kh + (size_t)bh * L_SEQ * D_QKV;
  const int m15 = lane & 15;
  const int kb  = (lane >> 4) << 1;
  const float* Arow = Q + (size_t)(tr * 16 + m15) * D_QKV;
  const float* Brow = K + (size_t)(tc * 16 + m15) * D_QKV;  // B[k][n] = K[n][k]

  v8f acc = {0.f, 0.f, 0.f, 0.f, 0.f, 0.f, 0.f, 0.f};
  for (int k0 = 0; k0 < D_QKV; k0 += 4) {
    v2f a;  a.x  = Arow[k0 + kb];  a.y  = Arow[k0 + kb + 1];
    v2f bm; bm.x = Brow[k0 + kb];  bm.y = Brow[k0 + kb + 1];
    acc = wmma4(a, bm, acc);
  }

  const float scale = 0.125f;  // 1/sqrt(D_QKV)
  float* Orow = attn + (size_t)bh * L_SEQ * L_SEQ;
#pragma unroll
  for (int r = 0; r < 8; ++r) {
    const int m = r + ((lane >> 4) << 3);
    Orow[(size_t)(tr * 16 + m) * L_SEQ + tc * 16 + m15] = acc[r] * scale;
  }
}

// ---------------------------------------------------------------------------
// 3) softmax over last axis, one wave32 per row (4096 rows). Applies mask
//    (mask==1 -> -inf, else subtract mask). Writes normalized attn in place
//    (d_out) and a transposed copy attnT[bh][k][q] for coalesced scan reads.
// ---------------------------------------------------------------------------
__global__ void softmax_kernel(float* __restrict__ attn,
                               const float* __restrict__ mask,
                               float* __restrict__ attnT) {
  const int wave = (blockIdx.x * blockDim.x + threadIdx.x) >> 5;
  const int lane = threadIdx.x & 31;
  const int q  = wave & (L_SEQ - 1);
  const int bh = wave >> 8;          // 0..15
  const int b  = bh >> 3;

  const float* mrow = mask + (size_t)(b * L_SEQ + q) * L_SEQ;  // mask (B,1,L,L)
  float* arow = attn + (size_t)wave * L_SEQ;

  float vals[8];
  float vmax = -INFINITY;
#pragma unroll
  for (int i = 0; i < 8; ++i) {
    const int kidx = lane + i * 32;
    const float mv = mrow[kidx];
    const float lg = arow[kidx];
    const float v  = (mv == 1.0f) ? -INFINITY : (lg - mv);
    vals[i] = v;
    vmax = fmaxf(vmax, v);
  }
#pragma unroll
  for (int off = 16; off > 0; off >>= 1)
    vmax = fmaxf(vmax, __shfl_xor(vmax, off, 32));

  float sum = 0.f;
#pragma unroll
  for (int i = 0; i < 8; ++i) { vals[i] = __expf(vals[i] - vmax); sum += vals[i]; }
#pragma unroll
  for (int off = 16; off > 0; off >>= 1)
    sum += __shfl_xor(sum, off, 32);

  const float inv = 1.0f / sum;
#pragma unroll
  for (int i = 0; i < 8; ++i) {
    const int kidx = lane + i * 32;
    const float a = vals[i] * inv;
    arow[kidx] = a;
    attnT[((size_t)bh * L_SEQ + kidx) * L_SEQ + q] = a;
  }
}

// ---------------------------------------------------------------------------
// 4) vW = v_h[b,h] (256x64) @ Win_{re,im}[h] (64x64).  Rank-1 factorization:
//    (attn_col ⊗ v_t) @ Win == attn_col ⊗ (v_t @ Win), replacing the
//    reference's 137 GFLOP of in-scan GEMMs with one 0.07 GFLOP precompute.
//    Output INTERLEAVED: vwri[((bh*256+t)*64 + e)*2 + {0=re,1=im}], so one
//    t-row is a contiguous 512 B block = one wave32 async B128 transfer.
//    16 bh * 2 (re/im) * 16 * 4 tiles = 2048 waves.
// ---------------------------------------------------------------------------
__global__ void vw_kernel(const float* __restrict__ vh,
                          const float* __restrict__ Wre,
                          const float* __restrict__ Wim,
                          float* __restrict__ vwri) {
  const int wave = (blockIdx.x * blockDim.x + threadIdx.x) >> 5;
  const int lane = threadIdx.x & 31;
  int tmp = wave;
  const int tc = tmp & 3;   tmp >>= 2;
  const int tr = tmp & 15;  tmp >>= 4;
  const int ri = tmp & 1;   tmp >>= 1;
  const int bh = tmp;                   // 0..15
  const int h  = bh & (NUM_HEADS - 1);

  const float* V = vh + (size_t)bh * L_SEQ * D_QKV;
  const float* W = ((ri == 0) ? Wre : Wim) + (size_t)h * D_QKV * D_QKV;
  float* OUT     = vwri + (size_t)bh * L_SEQ * VW_ROW + ri;

  const int m15 = lane & 15;
  const int kb  = (lane >> 4) << 1;
  const float* Arow = V + (size_t)(tr * 16 + m15) * D_QKV;

  v8f acc = {0.f, 0.f, 0.f, 0.f, 0.f, 0.f, 0.f, 0.f};
  for (int k0 = 0; k0 < D_QKV; k0 += 4) {
    v2f a;  a.x  = Arow[k0 + kb];  a.y = Arow[k0 + kb + 1];
    v2f bm; bm.x = W[(size_t)(k0 + kb)     * D_QKV + tc * 16 + m15];
            bm.y = W[(size_t)(k0 + kb + 1) * D_QKV + tc * 16 + m15];
    acc = wmma4(a, bm, acc);
  }
#pragma unroll
  for (int r = 0; r < 8; ++r) {
    const int m = r + ((lane >> 4) << 3);
    OUT[((size_t)(tr * 16 + m) * D_QKV + tc * 16 + m15) * 2] = acc[r];
  }
}

// ---------------------------------------------------------------------------
// 5) EUNN scan. One block per (b,h), one thread per query row q.
//    64-element complex state lives entirely in VGPRs (unrolled loops).
//    The shared per-step vW row (128 f32, interleaved re/im) is DOUBLE-
//    BUFFERED in LDS via async global->LDS DMA: while step t computes out of
//    vwbuf[t&1], wave 0 issues one GLOBAL_LOAD_ASYNC_TO_LDS_B128 (32 lanes *
//    16 B) for step t+1 into vwbuf[(t+1)&1], hiding the global latency on the
//    256-step sequential critical path. ASYNCcnt + barrier sequence:
//      wait_async0 (wave0) -> __syncthreads -> issue t+1 -> compute t
// ---------------------------------------------------------------------------
__global__ void scan_kernel(const float* __restrict__ attnT,
                            const float* __restrict__ vwri,
                            const float* __restrict__ theta,
                            const float* __restrict__ phi,
                            const float* __restrict__ rnn_bias,
                            float* __restrict__ rnn_out) {
  const int bh = blockIdx.x;
  const int h  = bh & (NUM_HEADS - 1);
  const int b  = bh >> 3;
  const int q  = threadIdx.x;          // 0..255
  const int tid = threadIdx.x;

  __shared__ float cs0[32], sn0[32], pr0[32], pi0[32];
  __shared__ float cs1[32], sn1[32], pr1[32], pi1[32];
  __shared__ float sb[D_QKV];
  __shared__ float vwbuf[2][VW_ROW];   // double-buffered interleaved vW row

  if (tid < 64) {
    const int c = tid >> 5, i = tid & 31;
    const float th = theta[((size_t)h * 2 + c) * 32 + i];
    const float ph = phi[((size_t)h * 2 + c) * 32 + i];
    if (c == 0) { cs0[i] = __cosf(th); sn0[i] = __sinf(th);
                  pr0[i] = __cosf(ph); pi0[i] = __sinf(ph); }
    else        { cs1[i] = __cosf(th); sn1[i] = __sinf(th);
                  pr1[i] = __cosf(ph); pi1[i] = __sinf(ph); }
  } else if (tid < 128) {
    const int e = tid - 64;
    sb[e] = rnn_bias[(size_t)h * D_QKV + e];
  }

  float hre[D_QKV], him[D_QKV];
#pragma unroll
  for (int e = 0; e < D_QKV; ++e) { hre[e] = 0.f; him[e] = 0.f; }

  const float* vw_base = vwri + (size_t)bh * L_SEQ * VW_ROW;
  const float* at_base = attnT + (size_t)bh * L_SEQ * L_SEQ;

  // Prologue: wave 0 prefetches row 0 into buffer 0 (lane l moves 16 B).
  if (tid < 32)
    async_load_lds_b128(vw_base + (tid << 2), &vwbuf[0][tid << 2]);

  for (int t = 0; t < L_SEQ; ++t) {
    if (tid < 32) wait_async0();   // row t landed in LDS (wave0's ASYNCcnt)
    __syncthreads();               // publish row t; all reads of buf[(t+1)&1] done
    if (tid < 32 && (t + 1) < L_SEQ)
      async_load_lds_b128(vw_base + (size_t)(t + 1) * VW_ROW + (tid << 2),
                          &vwbuf[(t + 1) & 1][tid << 2]);

    const float a = at_base[(size_t)t * L_SEQ + q];
    const float* vw = vwbuf[t & 1];

    // Unitary stage c=0: pairs (2i, 2i+1)
#pragma unroll
    for (int i = 0; i < 32; ++i) {
      const float ar = hre[2 * i],     ai = him[2 * i];
      const float br = hre[2 * i + 1], bi = him[2 * i + 1];
      const float tr_ = pr0[i] * br - pi0[i] * bi;   // ph * b
      const float ti_ = pr0[i] * bi + pi0[i] * br;
      const float nar = cs0[i] * ar - sn0[i] * tr_;
      const float nai = cs0[i] * ai - sn0[i] * ti_;
      const float ur  = pr0[i] * ar + pi0[i] * ai;   // conj(ph) * a
      const float ui  = pr0[i] * ai - pi0[i] * ar;
      const float nbr = sn0[i] * ur + cs0[i] * br;
      const float nbi = sn0[i] * ui + cs0[i] * bi;
      hre[2 * i] = nar;     him[2 * i] = nai;
      hre[2 * i + 1] = nbr; him[2 * i + 1] = nbi;
    }
    // Unitary stage c=1: roll(-1) pairing -> pairs (2i+1, (2i+2) mod 64)
#pragma unroll
    for (int i = 0; i < 32; ++i) {
      const int ia = 2 * i + 1, ib = (2 * i + 2) & (D_QKV - 1);
      const float ar = hre[ia], ai = him[ia];
      const float br = hre[ib], bi = him[ib];
      const float tr_ = pr1[i] * br - pi1[i] * bi;
      const float ti_ = pr1[i] * bi + pi1[i] * br;
      const float nar = cs1[i] * ar - sn1[i] * tr_;
      const float nai = cs1[i] * ai - sn1[i] * ti_;
      const float ur  = pr1[i] * ar + pi1[i] * ai;
      const float ui  = pr1[i] * ai - pi1[i] * ar;
      const float nbr = sn1[i] * ur + cs1[i] * br;
      const float nbi = sn1[i] * ui + cs1[i] * bi;
      hre[ia] = nar; him[ia] = nai;
      hre[ib] = nbr; him[ib] = nbi;
    }
    // rank-1 wx add + modrelu
#pragma unroll
    for (int e = 0; e < D_QKV; ++e) {
      const float zr = hre[e] + a * vw[2 * e];
      const float zi = him[e] + a * vw[2 * e + 1];
      const float m  = sqrtf(zr * zr + zi * zi);
      const float s  = fmaxf(m + sb[e], 0.f) / (m + 1e-5f);
      hre[e] = zr * s; him[e] = zi * s;
    }
  }

  // real(hT), written in (B, L, H*D) layout for the output projection
  float* orow = rnn_out + ((size_t)(b * L_SEQ + q)) * D_MODEL + h * D_QKV;
#pragma unroll
  for (int e = 0; e < D_QKV; ++e) orow[e] = hre[e];
}

// ---------------------------------------------------------------------------
// 6) out = rnn_out (512x512) @ Wo (512x512) + bo.  32*32 = 1024 wave tiles.
// ---------------------------------------------------------------------------
__global__ void gemm_out_kernel(const float* __restrict__ rnn,
                                const float* __restrict__ Wo,
                                const float* __restrict__ bo,
                                float* __restrict__ out) {
  const int wave = (blockIdx.x * blockDim.x + threadIdx.x) >> 5;
  const int lane = threadIdx.x & 31;
  const int tr = wave / 32, tc = wave % 32;

  const int m15 = lane & 15;
  const int kb  = (lane >> 4) << 1;
  const int rowBase = tr * 16, colBase = tc * 16;
  const float* Arow = rnn + (size_t)(rowBase + m15) * D_MODEL;

  v8f acc = {0.f, 0.f, 0.f, 0.f, 0.f, 0.f, 0.f, 0.f};
  for (int k0 = 0; k0 < D_MODEL; k0 += 4) {
    v2f a;  a.x  = Arow[k0 + kb];
            a.y  = Arow[k0 + kb + 1];
    v2f bm; bm.x = Wo[(size_t)(k0 + kb)     * D_MODEL + colBase + m15];
            bm.y = Wo[(size_t)(k0 + kb + 1) * D_MODEL + colBase + m15];
    acc = wmma4(a, bm, acc);
  }
  const float bval = bo[colBase + m15];
#pragma unroll
  for (int r = 0; r < 8; ++r) {
    const int m = r + ((lane >> 4) << 3);
    out[(size_t)(rowBase + m) * D_MODEL + colBase + m15] = acc[r] + bval;
  }
}

// ---------------------------------------------------------------------------
extern "C" void kernel_launch(void* const* d_in, const int* in_sizes, int n_in,
                              void* d_out, int out_size, void* d_ws, size_t ws_size,
                              hipStream_t stream) {
  const float* x_q     = (const float*)d_in[0];
  const float* x_k     = (const float*)d_in[1];
  const float* x_v     = (const float*)d_in[2];
  const float* mask    = (const float*)d_in[3];
  const float* Wq      = (const float*)d_in[4];
  const float* bq      = (const float*)d_in[5];
  const float* Wk      = (const float*)d_in[6];
  const float* bk      = (const float*)d_in[7];
  const float* Wv      = (const float*)d_in[8];
  const float* bv      = (const float*)d_in[9];
  const float* Wo      = (const float*)d_in[10];
  const float* bo      = (const float*)d_in[11];
  const float* theta   = (const float*)d_in[12];
  const float* phi     = (const float*)d_in[13];
  const float* Win_re  = (const float*)d_in[14];
  const float* Win_im  = (const float*)d_in[15];
  const float* rnn_b   = (const float*)d_in[16];

  // d_out = [ out (B,L,512) | attn (B,H,L,L) ]
  float* out_p  = (float*)d_out;
  float* attn_p = out_p + (size_t)ROWS * D_MODEL;   // 262144 offset

  // Workspace layout (floats): qh, kh, vh (262144 each), vwri (524288),
  // attnT (1048576), rnn (262144)  -> ~10 MB total
  float* ws    = (float*)d_ws;
  float* qh    = ws;
  float* kh    = qh + (size_t)BH * L_SEQ * D_QKV;
  float* vh    = kh + (size_t)BH * L_SEQ * D_QKV;
  float* vwri  = vh + (size_t)BH * L_SEQ * D_QKV;
  float* attnT = vwri + (size_t)BH * L_SEQ * VW_ROW;
  float* rnn   = attnT + (size_t)BH * L_SEQ * L_SEQ;

  gemm_qkv_kernel<<<384, 256, 0, stream>>>(x_q, x_k, x_v, Wq, Wk, Wv,
                                           bq, bk, bv, qh, kh, vh);
  logits_kernel<<<512, 256, 0, stream>>>(qh, kh, attn_p);
  softmax_kernel<<<512, 256, 0, stream>>>(attn_p, mask, attnT);
  vw_kernel<<<256, 256, 0, stream>>>(vh, Win_re, Win_im, vwri);
  scan_kernel<<<16, 256, 0, stream>>>(attnT, vwri, theta, phi, rnn_b, rnn);
  gemm_out_kernel<<<128, 256, 0, stream>>>(rnn, Wo, bo, out_p);
}